// SymEig_78838419685959
// MI455X (gfx1250) — compile-verified
//
#include <hip/hip_runtime.h>
#include <math.h>

#define TILE 256
#define EPS_NOISE 1e-6f

// ---- gfx1250 async global<->LDS paths (probe via __has_builtin) ------------
#if defined(__has_builtin)
#if __has_builtin(__builtin_amdgcn_global_load_async_to_lds_b32)
#define HAS_ASYNC_LD 1
#endif
#if __has_builtin(__builtin_amdgcn_global_store_async_from_lds_b32)
#define HAS_ASYNC_ST 1
#endif
#if __has_builtin(__builtin_amdgcn_s_wait_asynccnt)
#define HAS_WAIT_ASYNC_BUILTIN 1
#endif
#endif

typedef __attribute__((address_space(1))) int* gptr_i32;
typedef __attribute__((address_space(3))) int* lptr_i32;

__device__ __forceinline__ void wait_async_zero() {
#if defined(HAS_WAIT_ASYNC_BUILTIN)
    __builtin_amdgcn_s_wait_asynccnt(0);
#else
    asm volatile("s_wait_asynccnt 0" ::: "memory");
#endif
}

// ---- deterministic jitter (PCG-style hash -> uniform, unit variance) -------
__device__ __forceinline__ unsigned int hash_u32(unsigned int v) {
    v = v * 747796405u + 2891336453u;
    unsigned int w = ((v >> ((v >> 28u) + 4u)) ^ v) * 277803737u;
    return (w >> 22u) ^ w;
}

__device__ __forceinline__ float jitter(unsigned int seed) {
    float u = (float)(hash_u32(seed) >> 8) * (1.0f / 16777216.0f) - 0.5f;
    return u * 3.4641016f * EPS_NOISE;  // sqrt(12): unit-variance uniform
}

// ---- small vec helpers -----------------------------------------------------
__device__ __forceinline__ void cross3(const float a[3], const float b[3], float c[3]) {
    c[0] = a[1] * b[2] - a[2] * b[1];
    c[1] = a[2] * b[0] - a[0] * b[2];
    c[2] = a[0] * b[1] - a[1] * b[0];
}
__device__ __forceinline__ float dot3(const float a[3], const float b[3]) {
    return a[0] * b[0] + a[1] * b[1] + a[2] * b[2];
}
__device__ __forceinline__ void norm3(float v[3]) {
    float n = dot3(v, v);
    float r = rsqrtf(fmaxf(n, 1e-36f));
    v[0] *= r; v[1] *= r; v[2] *= r;
}

// Eigenvector of symmetric A for eigenvalue lam: best cross-product of rows of A-lam*I
__device__ __forceinline__ void eigvec_of(float a00, float a11, float a22,
                                          float a01, float a02, float a12,
                                          float lam, float v[3]) {
    float r0[3] = {a00 - lam, a01, a02};
    float r1[3] = {a01, a11 - lam, a12};
    float r2[3] = {a02, a12, a22 - lam};
    float c0[3], c1[3], c2[3];
    cross3(r0, r1, c0);
    cross3(r0, r2, c1);
    cross3(r1, r2, c2);
    float n0 = dot3(c0, c0), n1 = dot3(c1, c1), n2 = dot3(c2, c2);
    float nmax = n0;
    const float* cb = c0;
    if (n1 > nmax) { nmax = n1; cb = c1; }
    if (n2 > nmax) { nmax = n2; cb = c2; }
    if (nmax < 1e-36f) { v[0] = 1.f; v[1] = 0.f; v[2] = 0.f; return; }
    float rinv = rsqrtf(nmax);
    v[0] = cb[0] * rinv; v[1] = cb[1] * rinv; v[2] = cb[2] * rinv;
}

// ---------------------------------------------------------------------------
__global__ void __launch_bounds__(TILE)
symeig3x3_kernel(const float* __restrict__ x, float* __restrict__ out, int B) {
    __shared__ __align__(16) float sIn[TILE * 9];
    __shared__ __align__(16) float sVec[TILE * 9];
    __shared__ __align__(16) float sVal[TILE * 3];

    const int t = threadIdx.x;
    const int blk = blockIdx.x;
    const int inBase = blk * (TILE * 9);
    const int totalIn = B * 9;
    const bool full = (blk + 1) * TILE <= B;  // uniform per block

    // ---- stage input tile into LDS, fully coalesced -----------------------
#if defined(HAS_ASYNC_LD)
    if (full) {
        // Shared VGPR base, literal IOFFSET advances both LDS and global
        // addresses by 1024 B per step (ISA: INST_OFFSET added to both).
        const float* gb = x + inBase + t;
        lptr_i32 lb = (lptr_i32)(&sIn[t]);
#define LD_STEP(OFF) __builtin_amdgcn_global_load_async_to_lds_b32((gptr_i32)gb, lb, (OFF), 0)
        LD_STEP(0);    LD_STEP(1024); LD_STEP(2048);
        LD_STEP(3072); LD_STEP(4096); LD_STEP(5120);
        LD_STEP(6144); LD_STEP(7168); LD_STEP(8192);
#undef LD_STEP
    } else {
#pragma unroll
        for (int i = 0; i < 9; ++i) {
            int g = inBase + t + i * TILE;
            if (g < totalIn) {
                __builtin_amdgcn_global_load_async_to_lds_b32(
                    (gptr_i32)(x + g), (lptr_i32)(&sIn[t + i * TILE]), 0, 0);
            }
        }
    }
    wait_async_zero();
#else
#pragma unroll
    for (int i = 0; i < 9; ++i) {
        int g = inBase + t + i * TILE;
        if (g < totalIn) sIn[t + i * TILE] = x[g];
    }
#endif
    __syncthreads();

    // ---- per-thread analytic 3x3 symmetric eig ----------------------------
    const int b = blk * TILE + t;
    if (b < B) {
        const float* m = &sIn[t * 9];
        float a00 = m[0], a01m = m[1], a02m = m[2];
        float a10 = m[3], a11 = m[4], a12m = m[5];
        float a20 = m[6], a21 = m[7], a22 = m[8];

        // deterministic jitter (stands in for eps * N(0,1); breaks degeneracy)
        unsigned int s = (unsigned int)b * 9u;
        a00 += jitter(s + 0u); a01m += jitter(s + 1u); a02m += jitter(s + 2u);
        a10 += jitter(s + 3u); a11 += jitter(s + 4u); a12m += jitter(s + 5u);
        a20 += jitter(s + 6u); a21 += jitter(s + 7u); a22 += jitter(s + 8u);

        // symmetrize
        float a01 = 0.5f * (a01m + a10);
        float a02 = 0.5f * (a02m + a20);
        float a12 = 0.5f * (a12m + a21);

        // Cardano eigenvalues (ascending: w0 <= w1 <= w2)
        float q = (a00 + a11 + a22) * (1.0f / 3.0f);
        float b00 = a00 - q, b11 = a11 - q, b22 = a22 - q;
        float p1 = a01 * a01 + a02 * a02 + a12 * a12;
        float p2 = b00 * b00 + b11 * b11 + b22 * b22 + 2.0f * p1;
        float p = sqrtf(p2 * (1.0f / 6.0f)) + 1e-30f;
        float ip = 1.0f / p;
        float c00 = b00 * ip, c11 = b11 * ip, c22 = b22 * ip;
        float c01 = a01 * ip, c02 = a02 * ip, c12 = a12 * ip;
        float detB = c00 * (c11 * c22 - c12 * c12)
                   - c01 * (c01 * c22 - c12 * c02)
                   + c02 * (c01 * c12 - c11 * c02);
        float r = fminf(fmaxf(0.5f * detB, -1.0f), 1.0f);
        float phi = acosf(r) * (1.0f / 3.0f);
        float twop = 2.0f * p;
        float w2 = q + twop * cosf(phi);                       // largest
        float w0 = q + twop * cosf(phi + 2.0943951023931953f); // smallest
        float w1 = 3.0f * q - w2 - w0;                         // middle

        // Eigenvectors (orthonormal columns, ascending order)
        float v0[3], v1[3], v2[3];
        eigvec_of(a00, a11, a22, a01, a02, a12, w0, v0);
        eigvec_of(a00, a11, a22, a01, a02, a12, w2, v2);
        float d02 = dot3(v2, v0);
        v2[0] -= d02 * v0[0]; v2[1] -= d02 * v0[1]; v2[2] -= d02 * v0[2];
        norm3(v2);
        cross3(v2, v0, v1);  // orthogonal to both, unit length

        sVal[t * 3 + 0] = w0;
        sVal[t * 3 + 1] = w1;
        sVal[t * 3 + 2] = w2;
#pragma unroll
        for (int rr = 0; rr < 3; ++rr) {
            sVec[t * 9 + rr * 3 + 0] = v0[rr];
            sVec[t * 9 + rr * 3 + 1] = v1[rr];
            sVec[t * 9 + rr * 3 + 2] = v2[rr];
        }
    }
    __syncthreads();

    // ---- coalesced stores: eigenvalues [B*3], then eigenvectors [B*9] -----
    const int evalBase = blk * (TILE * 3);
    const int evecBase = 3 * B + blk * (TILE * 9);

#if defined(HAS_ASYNC_ST)
    if (full) {
        float* gv = out + evalBase + t;
        lptr_i32 lv = (lptr_i32)(&sVal[t]);
#define ST_STEP(G, L, OFF) \
    __builtin_amdgcn_global_store_async_from_lds_b32((gptr_i32)(G), (L), (OFF), 0)
        ST_STEP(gv, lv, 0); ST_STEP(gv, lv, 1024); ST_STEP(gv, lv, 2048);
        float* ge = out + evecBase + t;
        lptr_i32 le = (lptr_i32)(&sVec[t]);
        ST_STEP(ge, le, 0);    ST_STEP(ge, le, 1024); ST_STEP(ge, le, 2048);
        ST_STEP(ge, le, 3072); ST_STEP(ge, le, 4096); ST_STEP(ge, le, 5120);
        ST_STEP(ge, le, 6144); ST_STEP(ge, le, 7168); ST_STEP(ge, le, 8192);
#undef ST_STEP
        // s_endpgm performs an implicit wait-idle; LDS is not reused after this.
        return;
    }
#endif
    {
        int nb = B - blk * TILE;
        if (nb > TILE) nb = TILE;
#pragma unroll
        for (int i = 0; i < 3; ++i) {
            int j = t + i * TILE;
            if (j < nb * 3) out[evalBase + j] = sVal[j];
        }
#pragma unroll
        for (int i = 0; i < 9; ++i) {
            int j = t + i * TILE;
            if (j < nb * 9) out[evecBase + j] = sVec[j];
        }
    }
}

extern "C" void kernel_launch(void* const* d_in, const int* in_sizes, int n_in,
                              void* d_out, int out_size, void* d_ws, size_t ws_size,
                              hipStream_t stream) {
    (void)n_in; (void)out_size; (void)d_ws; (void)ws_size;
    const float* x = (const float*)d_in[0];
    const int B = in_sizes[0] / 9;  // [B,3,3] float32
    float* out = (float*)d_out;     // B*3 eigenvalues, then B*9 eigenvectors
    const int grid = (B + TILE - 1) / TILE;
    hipLaunchKernelGGL(symeig3x3_kernel, dim3(grid), dim3(TILE), 0, stream,
                       x, out, B);
}